// Infusion_4294967296641
// MI455X (gfx1250) — compile-verified
//
#include <hip/hip_runtime.h>

// Dilated-window attention via dilation-lattice decomposition:
// each (x%3, y%3) residue subgrid is a dense ksize=5 neighborhood attention.
// Per 4x4 query tile: S(16x64)=Q(16x32)@Kunion(32x64), softmax->W (zeros off
// window), O(16x32)=W(16x64)@Vunion(64x32).  All fp32 via v_wmma_f32_16x16x4_f32.
// OOB keys/values are handled by clamped unconditional loads * {0,1} mask, so
// there are no exec-mask diamonds around the global loads.

typedef __attribute__((ext_vector_type(2))) float v2f;
typedef __attribute__((ext_vector_type(8))) float v8f;

#define HW   16384   // 128*128
#define IMG  128
#define SROW 66      // LDS row stride (even -> aligned pair reads, conflict-free)

__global__ __launch_bounds__(256) void infusion_wmma_kernel(
    const float* __restrict__ Vin, const float* __restrict__ Kin,
    const float* __restrict__ Qin, float* __restrict__ out)
{
  int bid = blockIdx.x;
  const int tj = bid % 11; bid /= 11;
  const int ti = bid % 11; bid /= 11;
  const int ry = bid % 3;  bid /= 3;
  const int rx = bid % 3;  bid /= 3;
  const int n  = bid;                       // batch 0..7

  const int Hs = (130 - rx) / 3;            // subgrid extents (43/43/42)
  const int Ws = (130 - ry) / 3;

  __shared__ float smem[8][16 * SROW];      // per-head scores -> weights
  __shared__ int   s_ubase[64];             // union key: clamped pixel offset
  __shared__ float s_umask[64];             // union key: 1.0 valid / 0.0 OOB

  // Union-window geometry depends only on the tile: compute once per block.
  if (threadIdx.x < 64) {
    const int u = threadIdx.x, ui = u >> 3, uj = u & 7;
    const int ski = ti * 4 + ui - 2, skj = tj * 4 + uj - 2;
    const bool ok = (ski >= 0) && (ski < Hs) && (skj >= 0) && (skj < Ws);
    s_ubase[u] = ok ? ((rx + 3 * ski) * IMG + (ry + 3 * skj)) : 0;
    s_umask[u] = ok ? 1.0f : 0.0f;
  }
  __syncthreads();

  const int lane   = threadIdx.x & 31;
  const int h      = threadIdx.x >> 5;      // wave == head (wave32)
  const int l15    = lane & 15;
  const int hiHalf = lane >> 4;
  const int cofs   = hiHalf * 2;            // lane's offset inside a K=4 chunk

  float* S = smem[h];

  const size_t headOff = ((size_t)n * 256 + (size_t)h * 32) * HW;
  const float* Qh = Qin + headOff;
  const float* Kh = Kin + headOff;
  const float* Vh = Vin + headOff;

  // ---- query (A row) owned by this lane; clamp OOB (row is never stored) ---
  const int qm  = l15;
  const int qi  = qm >> 2, qj = qm & 3;
  const int sqi = ti * 4 + qi, sqj = tj * 4 + qj;
  const bool qok = (sqi < Hs) && (sqj < Ws);
  const int qbase = qok ? ((rx + 3 * sqi) * IMG + (ry + 3 * sqj)) : 0;
  const float* Qp = Qh + (size_t)cofs * HW + qbase;

  // ---- phase-1 B bases: loop-invariant pointers + masks ----
  const float* Kp[4];
  float km[4];
#pragma unroll
  for (int t = 0; t < 4; ++t) {
    const int u = t * 16 + l15;
    Kp[t] = Kh + (size_t)cofs * HW + s_ubase[u];
    km[t] = s_umask[u];
  }

  // ================= Phase 1: S = Q @ K_union ===============================
  v8f acc[4] = {v8f{}, v8f{}, v8f{}, v8f{}};
#pragma unroll
  for (int c0 = 0; c0 < 32; c0 += 4) {
    v2f a;
    a.x = Qp[(size_t)c0 * HW];
    a.y = Qp[(size_t)(c0 + 1) * HW];
#pragma unroll
    for (int t = 0; t < 4; ++t) {
      v2f b;
      b.x = km[t] * Kp[t][(size_t)c0 * HW];
      b.y = km[t] * Kp[t][(size_t)(c0 + 1) * HW];
      acc[t] = __builtin_amdgcn_wmma_f32_16x16x4_f32(
          false, a, false, b, (short)0, acc[t], false, false);
    }
  }

  // spill S (C layout: row = r + 8*hiHalf, col = t*16 + l15)
#pragma unroll
  for (int t = 0; t < 4; ++t)
#pragma unroll
    for (int r = 0; r < 8; ++r)
      S[(r + hiHalf * 8) * SROW + t * 16 + l15] = acc[t][r];
  __syncthreads();

  // ============== softmax over the 25 window taps, S -> W in place ==========
  if (lane < 16) {
    float ev[25];
    float mx = -3.0e38f;
#pragma unroll
    for (int fi = 0; fi < 5; ++fi)
#pragma unroll
      for (int fj = 0; fj < 5; ++fj)
        mx = fmaxf(mx, S[qm * SROW + (qi + fi) * 8 + (qj + fj)]);
    float sum = 0.0f;
#pragma unroll
    for (int fi = 0; fi < 5; ++fi)
#pragma unroll
      for (int fj = 0; fj < 5; ++fj) {
        float e = expf(S[qm * SROW + (qi + fi) * 8 + (qj + fj)] - mx);
        ev[fi * 5 + fj] = e;
        sum += e;
      }
    const float inv = 1.0f / sum;
#pragma unroll
    for (int u = 0; u < 64; ++u) {
      int fi = (u >> 3) - qi, fj = (u & 7) - qj;
      float w = 0.0f;
      if (fi >= 0 && fi < 5 && fj >= 0 && fj < 5) w = ev[fi * 5 + fj] * inv;
      S[qm * SROW + u] = w;
    }
  }
  __syncthreads();

  // ================= Phase 2: O = W @ V_union ===============================
  const float* Vp0 = Vh + (size_t)l15 * HW;        // N-tile 0: channel l15
  const float* Vp1 = Vp0 + (size_t)16 * HW;        // N-tile 1: channel 16+l15
  v8f oacc[2] = {v8f{}, v8f{}};
#pragma unroll
  for (int c = 0; c < 16; ++c) {
    const int u1 = 4 * c + cofs;
    v2f a;                                          // A = W chunk (LDS)
    a.x = S[qm * SROW + u1];
    a.y = S[qm * SROW + u1 + 1];

    const int   b1 = s_ubase[u1],     b2 = s_ubase[u1 + 1];
    const float m1 = s_umask[u1],     m2 = s_umask[u1 + 1];

    v2f b;
    b.x = m1 * Vp0[b1];
    b.y = m2 * Vp0[b2];
    oacc[0] = __builtin_amdgcn_wmma_f32_16x16x4_f32(
        false, a, false, b, (short)0, oacc[0], false, false);
    b.x = m1 * Vp1[b1];
    b.y = m2 * Vp1[b2];
    oacc[1] = __builtin_amdgcn_wmma_f32_16x16x4_f32(
        false, a, false, b, (short)0, oacc[1], false, false);
  }

  // ================= store (C layout) =======================================
#pragma unroll
  for (int r = 0; r < 8; ++r) {
    const int m   = r + hiHalf * 8;
    const int smi = ti * 4 + (m >> 2), smj = tj * 4 + (m & 3);
    if (smi < Hs && smj < Ws) {
      const size_t pix = (size_t)(rx + 3 * smi) * IMG + (ry + 3 * smj);
#pragma unroll
      for (int nt = 0; nt < 2; ++nt) {
        const size_t ch = (size_t)h * 32 + nt * 16 + l15;
        out[((size_t)n * 256 + ch) * HW + pix] = oacc[nt][r];
      }
    }
  }
}

extern "C" void kernel_launch(void* const* d_in, const int* in_sizes, int n_in,
                              void* d_out, int out_size, void* d_ws, size_t ws_size,
                              hipStream_t stream) {
  (void)in_sizes; (void)n_in; (void)out_size; (void)d_ws; (void)ws_size;
  const float* V = (const float*)d_in[0];
  const float* K = (const float*)d_in[1];
  const float* Q = (const float*)d_in[2];
  float* out = (float*)d_out;
  // ksize=5, dilation=3, n_heads=8 are fixed by the reference setup.
  const int grid = 8 * 3 * 3 * 11 * 11;     // batch x residues x 11x11 tiles
  infusion_wmma_kernel<<<grid, 256, 0, stream>>>(V, K, Q, out);
}